// NGU_6098853560364
// MI455X (gfx1250) — compile-verified
//
#include <hip/hip_runtime.h>

#define CAP 8192
#define NENV 64
#define DIM 256
#define TOPK 10
#define EPS_C 0.001f
#define MIN_DIST 0.008f
#define MAX_SIM 2.0f
#define C_CONST 1.0f
#define L_CLAMP 5.0f
#define BIGF 1e30f

typedef __attribute__((ext_vector_type(2))) float v2f;
typedef __attribute__((ext_vector_type(8))) float v8f;

// ---------------------------------------------------------------------------
// Bitonic helpers: merge two ascending 10-lists, keep the 10 smallest.
// All indices compile-time constant -> stays in VGPRs (no scratch).
// ---------------------------------------------------------------------------
__device__ __forceinline__ void sort16(float v[16]) {
#pragma unroll
  for (int k = 2; k <= 16; k <<= 1) {
#pragma unroll
    for (int j = k >> 1; j > 0; j >>= 1) {
#pragma unroll
      for (int i = 0; i < 16; ++i) {
        const int l = i ^ j;
        if (l > i) {
          const bool asc = ((i & k) == 0);
          const float x = v[i], y = v[l];
          const bool swap = asc ? (x > y) : (x < y);
          v[i] = swap ? y : x;
          v[l] = swap ? x : y;
        }
      }
    }
  }
}

__device__ __forceinline__ void merge10(float a[TOPK], const float b[TOPK]) {
  float c[16];
#pragma unroll
  for (int i = 0; i < TOPK; ++i) c[i] = fminf(a[i], b[TOPK - 1 - i]);
#pragma unroll
  for (int i = TOPK; i < 16; ++i) c[i] = BIGF;
  sort16(c);  // ascending; first 10 = smallest of the union
#pragma unroll
  for (int i = 0; i < TOPK; ++i) a[i] = c[i];
}

// ---------------------------------------------------------------------------
// Phase 1: grid (16 c-chunks, 64 envs), 256 threads (8 waves).
// Each block: 512 rows of one env. Per wave-tile of 16 rows:
//   accDot  += A(16x4 data) x B(obs replicated over 16 cols)
//   accGram += A x A   (same registers as A and B fragments -> diag = norms)
// Then per-block bitonic top-10 of the 512 distances.
// ---------------------------------------------------------------------------
__global__ __launch_bounds__(256) void ngu_phase1(
    const float* __restrict__ obs, const float* __restrict__ data,
    const int* __restrict__ nin_p, float* __restrict__ ws_part) {
  __shared__ __align__(16) float lds_obs[DIM];
  __shared__ float lds_di[512];
  __shared__ float lds_lists[256 * TOPK];

  const int env = blockIdx.y;
  const int cb = blockIdx.x;
  const int t = threadIdx.x;
  const int c_base = cb * 512;
  const int nin = nin_p[env];

  lds_obs[t] = obs[env * DIM + t];
  __syncthreads();

  // ||obs||^2 (uniform across block; cheap broadcast reads from LDS)
  float on2 = 0.f;
#pragma unroll 8
  for (int i = 0; i < DIM; ++i) on2 += lds_obs[i] * lds_obs[i];

  const int wid = t >> 5;
  const int lane = t & 31;
  const int half = lane >> 4;  // A/B fragment K half
  const int m = lane & 15;     // row within tile (A) / column (B, Gram)
  const int dbase = 2 * half;

#pragma unroll
  for (int tt = 0; tt < 4; ++tt) {
    const int tile = wid + tt * 8;        // 0..31
    const int rloc = tile * 16 + m;       // local row 0..511
    const int row = c_base + rloc;        // global c
    const float* aptr = data + ((size_t)row * (NENV * DIM) + (size_t)env * DIM);

    // prefetch next tile's row for this wave (clamped, speculative)
    {
      int pr = row + 128;
      if (pr > CAP - 1) pr = CAP - 1;
      __builtin_prefetch(data + ((size_t)pr * (NENV * DIM) + (size_t)env * DIM), 0, 0);
    }

    v8f accDot = {0.f, 0.f, 0.f, 0.f, 0.f, 0.f, 0.f, 0.f};
    v8f accGram = {0.f, 0.f, 0.f, 0.f, 0.f, 0.f, 0.f, 0.f};

#pragma unroll 4
    for (int kk = 0; kk < 64; ++kk) {
      const int d0 = 4 * kk + dbase;
      v2f a = *(const v2f*)(aptr + d0);        // A fragment (data rows)
      v2f b = *(const v2f*)(&lds_obs[d0]);     // B fragment (obs, all cols equal)
      // D = A*B + C : 16x16x4 f32 WMMA
      accDot = __builtin_amdgcn_wmma_f32_16x16x4_f32(
          false, a, false, b, (short)0, accDot, false, false);
      // Gram = A*A^T via identical fragment layout -> diagonal = row norms
      accGram = __builtin_amdgcn_wmma_f32_16x16x4_f32(
          false, a, false, a, (short)0, accGram, false, false);
    }

    // C/D layout: VGPR j holds D[M = j + 8*half][N = m]. Diagonal when m == M.
#pragma unroll
    for (int j = 0; j < 8; ++j) {
      const int r = j + 8 * half;
      if (m == r) {
        const float di = on2 + accGram[j] - 2.0f * accDot[j];
        const int cg = c_base + tile * 16 + r;
        lds_di[tile * 16 + r] = (cg < nin) ? di : BIGF;
      }
    }
  }
  __syncthreads();

  // per-thread sorted list of its 2 distances, padded with BIG
  float myl[TOPK];
  {
    const float d0 = lds_di[2 * t], d1 = lds_di[2 * t + 1];
    myl[0] = fminf(d0, d1);
    myl[1] = fmaxf(d0, d1);
#pragma unroll
    for (int i = 2; i < TOPK; ++i) myl[i] = BIGF;
  }
#pragma unroll
  for (int i = 0; i < TOPK; ++i) lds_lists[t * TOPK + i] = myl[i];

  for (int s = 128; s >= 1; s >>= 1) {
    __syncthreads();
    if (t < s) {
      float ob[TOPK];
#pragma unroll
      for (int i = 0; i < TOPK; ++i) ob[i] = lds_lists[(t + s) * TOPK + i];
      merge10(myl, ob);
#pragma unroll
      for (int i = 0; i < TOPK; ++i) lds_lists[t * TOPK + i] = myl[i];
    }
  }
  if (t == 0) {
    float* outp = ws_part + ((size_t)env * 16 + cb) * TOPK;
#pragma unroll
    for (int i = 0; i < TOPK; ++i) outp[i] = myl[i];
  }
}

// ---------------------------------------------------------------------------
// Phase 2: one block per env; merge 16 partial top-10 lists (160 candidates).
// ---------------------------------------------------------------------------
__global__ __launch_bounds__(256) void ngu_phase2(
    const float* __restrict__ ws_part, const int* __restrict__ nin_p,
    const int* __restrict__ kp, float* __restrict__ ws_dists,
    float* __restrict__ ws_maxd, float* __restrict__ ws_valid) {
  __shared__ float lds_lists[256 * TOPK];
  const int env = blockIdx.x;
  const int t = threadIdx.x;

  float myl[TOPK];
#pragma unroll
  for (int i = 0; i < TOPK; ++i) myl[i] = BIGF;
  if (t < 16 * TOPK) myl[0] = ws_part[(size_t)env * (16 * TOPK) + t];
#pragma unroll
  for (int i = 0; i < TOPK; ++i) lds_lists[t * TOPK + i] = myl[i];

  for (int s = 128; s >= 1; s >>= 1) {
    __syncthreads();
    if (t < s) {
      float ob[TOPK];
#pragma unroll
      for (int i = 0; i < TOPK; ++i) ob[i] = lds_lists[(t + s) * TOPK + i];
      merge10(myl, ob);
#pragma unroll
      for (int i = 0; i < TOPK; ++i) lds_lists[t * TOPK + i] = myl[i];
    }
  }

  if (t == 0) {
    const int kval = kp[0];
    const float valid = (nin_p[env] >= kval) ? 1.f : 0.f;
#pragma unroll
    for (int i = 0; i < TOPK; ++i) ws_dists[env * TOPK + i] = myl[i] * valid;
    ws_maxd[env] = myl[TOPK - 1] * valid;
    ws_valid[env] = valid;
  }
}

// ---------------------------------------------------------------------------
// Phase 3: single tiny block; cross-env average + kernel reward.
// ---------------------------------------------------------------------------
__global__ __launch_bounds__(64) void ngu_phase3(
    const float* __restrict__ ws_dists, const float* __restrict__ ws_maxd,
    const float* __restrict__ ws_valid, const float* __restrict__ r_rnd,
    float* __restrict__ out) {
  __shared__ float savg;
  const int e = threadIdx.x;
  if (e == 0) {
    float cnt = 0.f, s = 0.f;
    for (int i = 0; i < NENV; ++i) {
      cnt += ws_valid[i];
      s += ws_maxd[i];  // already multiplied by valid
    }
    savg = (cnt > 0.f) ? (s / fmaxf(cnt, 1.f)) : 0.f;
  }
  __syncthreads();
  const float avg = savg;
  const float denom = (avg > 1e-5f) ? avg : 1.f;

  float ks = 0.f;
#pragma unroll
  for (int i = 0; i < TOPK; ++i) {
    float d = ws_dists[e * TOPK + i] / denom;
    d = fmaxf(d - MIN_DIST, 0.f);
    ks += EPS_C / (d + EPS_C);
  }
  const float s = sqrtf(C_CONST + ks);
  const float r = (s > MAX_SIM) ? 0.f : (1.f / s);
  const float mod = fminf(fmaxf(r_rnd[e], 1.f), L_CLAMP);
  out[e] = r * mod;
}

// ---------------------------------------------------------------------------
extern "C" void kernel_launch(void* const* d_in, const int* in_sizes, int n_in,
                              void* d_out, int out_size, void* d_ws,
                              size_t ws_size, hipStream_t stream) {
  const float* obs = (const float*)d_in[0];    // (64, 256) f32
  const float* data = (const float*)d_in[1];   // (8192, 64, 256) f32
  const float* r_rnd = (const float*)d_in[2];  // (64,) f32
  const int* nin = (const int*)d_in[3];        // (64,) i32
  const int* kp = (const int*)d_in[4];         // scalar i32 (=10)
  float* out = (float*)d_out;                  // (64,) f32

  float* ws = (float*)d_ws;
  float* ws_part = ws;                               // 64*16*10 = 10240
  float* ws_dists = ws_part + (size_t)NENV * 16 * TOPK;  // 640
  float* ws_maxd = ws_dists + (size_t)NENV * TOPK;       // 64
  float* ws_valid = ws_maxd + NENV;                      // 64

  dim3 g1(16, NENV);
  ngu_phase1<<<g1, 256, 0, stream>>>(obs, data, nin, ws_part);
  ngu_phase2<<<NENV, 256, 0, stream>>>(ws_part, nin, kp, ws_dists, ws_maxd,
                                       ws_valid);
  ngu_phase3<<<1, 64, 0, stream>>>(ws_dists, ws_maxd, ws_valid, r_rnd, out);
}